// TGCNModel_20555713478796
// MI455X (gfx1250) — compile-verified
//
#include <hip/hip_runtime.h>
#include <math.h>

typedef __attribute__((ext_vector_type(2))) float v2f;
typedef __attribute__((ext_vector_type(8))) float v8f;

#define N_NODES 100000
#define FIN     96        // NODE_FEAT * TIME_STEPS
#define HID     128
#define N_EDGES 1600000

// ---------------- degree / rsqrt ----------------
__global__ __launch_bounds__(256) void k_init_deg(float* __restrict__ deg) {
  int i = blockIdx.x * 256 + threadIdx.x;
  if (i < N_NODES) deg[i] = 1.0f;              // self-loop
}

__global__ __launch_bounds__(256) void k_edge_deg(const int* __restrict__ dst,
                                                  float* __restrict__ deg) {
  int e = blockIdx.x * 256 + threadIdx.x;
  if (e < N_EDGES) atomicAdd(&deg[dst[e]], 1.0f);
}

__global__ __launch_bounds__(256) void k_rsqrt(float* __restrict__ deg) {
  int i = blockIdx.x * 256 + threadIdx.x;
  if (i < N_NODES) deg[i] = rsqrtf(deg[i]);
}

// ---------------- feature GEMM: H[N,128] = X[N,96] @ W[96,128] (fp32 WMMA) ----
// One wave -> one 16x16 output tile. Block = 8 waves = 16 rows x 128 cols.
__global__ __launch_bounds__(256) void k_feat_gemm(const float* __restrict__ X,
                                                   const float* __restrict__ W,
                                                   float* __restrict__ Hout) {
  const int wave = threadIdx.x >> 5;
  const int lane = threadIdx.x & 31;
  const int half = lane >> 4;                  // 0: K pair {0,1}, 1: K pair {2,3}
  const int l16  = lane & 15;
  const int row0 = blockIdx.x * 16;
  const int col  = wave * 16 + l16;

  v8f acc = {};
  const float* arow = X + (size_t)(row0 + l16) * FIN + half * 2;
#pragma unroll
  for (int k0 = 0; k0 < FIN; k0 += 4) {
    const int kb = k0 + half * 2;
    v2f a, b;
    a.x = arow[k0];
    a.y = arow[k0 + 1];
    b.x = W[(size_t)kb * HID + col];
    b.y = W[(size_t)(kb + 1) * HID + col];
    acc = __builtin_amdgcn_wmma_f32_16x16x4_f32(false, a, false, b,
                                                (short)0, acc, false, false);
  }
  // C/D layout: VGPR r -> row (r + 8*half), col = lane%16 (+16*wave)
  float* crow = Hout + (size_t)(row0 + half * 8) * HID + col;
#pragma unroll
  for (int r = 0; r < 8; ++r) crow[(size_t)r * HID] = acc[r];
}

// ---------------- agg init: agg = h*dinv^2 + bias ----------------
__global__ __launch_bounds__(256) void k_agg_init(const float* __restrict__ h,
                                                  const float* __restrict__ dinv,
                                                  const float* __restrict__ bias,
                                                  float* __restrict__ agg) {
  int i = blockIdx.x * 256 + threadIdx.x;      // over N*128
  int n = i >> 7, c = i & 127;
  float di = dinv[n];
  agg[i] = h[i] * di * di + bias[c];
}

// ---------------- edge scatter: agg[dst] += h[src]*dinv[src]*dinv[dst] -------
// One wave per edge: lane handles 4 contiguous floats (float4 gather, 4 atomics)
__global__ __launch_bounds__(256) void k_scatter(const int* __restrict__ src,
                                                 const int* __restrict__ dst,
                                                 const float* __restrict__ dinv,
                                                 const float* __restrict__ h,
                                                 float* __restrict__ agg) {
  const int wave = threadIdx.x >> 5;
  const int lane = threadIdx.x & 31;
  const int e = blockIdx.x * 8 + wave;
  const int s = src[e];
  const int d = dst[e];
  const float norm = dinv[s] * dinv[d];
  const float4 v = *reinterpret_cast<const float4*>(h + (size_t)s * HID + lane * 4);
  float* o = agg + (size_t)d * HID + lane * 4;
  atomicAdd(o + 0, v.x * norm);
  atomicAdd(o + 1, v.y * norm);
  atomicAdd(o + 2, v.z * norm);
  atomicAdd(o + 3, v.w * norm);
}

// ---------------- fused gate GEMMs + GRU combine + output projection ---------
// Z  = sigmoid(aggz @ Lz[0:128,:] + lbz)
// Ht = tanh   (aggh @ Lh[0:128,:] + lbh)
// out = ((1-Z)*Ht) @ Wout + bout
__global__ __launch_bounds__(256) void k_gate_out(const float* __restrict__ aggz,
                                                  const float* __restrict__ aggh,
                                                  const float* __restrict__ Lz,
                                                  const float* __restrict__ lbz,
                                                  const float* __restrict__ Lh,
                                                  const float* __restrict__ lbh,
                                                  const float* __restrict__ Wout,
                                                  const float* __restrict__ bout,
                                                  float* __restrict__ out) {
  __shared__ float hn[16 * HID];               // (1-Z)*Ht staging tile, 8KB
  const int wave = threadIdx.x >> 5;
  const int lane = threadIdx.x & 31;
  const int half = lane >> 4;
  const int l16  = lane & 15;
  const int row0 = blockIdx.x * 16;
  const int col  = wave * 16 + l16;

  const float bz = lbz[col];
  const float bh = lbh[col];
  v8f cz, ch;
#pragma unroll
  for (int r = 0; r < 8; ++r) { cz[r] = bz; ch[r] = bh; }

  const float* az = aggz + (size_t)(row0 + l16) * HID + half * 2;
  const float* ah = aggh + (size_t)(row0 + l16) * HID + half * 2;
#pragma unroll
  for (int k0 = 0; k0 < HID; k0 += 4) {
    const int kb = k0 + half * 2;
    v2f a, b;
    a.x = az[k0];
    a.y = az[k0 + 1];
    b.x = Lz[(size_t)kb * HID + col];
    b.y = Lz[(size_t)(kb + 1) * HID + col];
    cz = __builtin_amdgcn_wmma_f32_16x16x4_f32(false, a, false, b,
                                               (short)0, cz, false, false);
    a.x = ah[k0];
    a.y = ah[k0 + 1];
    b.x = Lh[(size_t)kb * HID + col];
    b.y = Lh[(size_t)(kb + 1) * HID + col];
    ch = __builtin_amdgcn_wmma_f32_16x16x4_f32(false, a, false, b,
                                               (short)0, ch, false, false);
  }

#pragma unroll
  for (int r = 0; r < 8; ++r) {
    float z = 1.0f / (1.0f + __expf(-cz[r]));
    float t = tanhf(ch[r]);
    hn[(r + half * 8) * HID + col] = (1.0f - z) * t;
  }
  __syncthreads();

  if (threadIdx.x < 16) {
    float acc = bout[0];
    const float* row = hn + threadIdx.x * HID;
#pragma unroll 8
    for (int c = 0; c < HID; ++c) acc = fmaf(row[c], Wout[c], acc);
    out[row0 + threadIdx.x] = acc;
  }
}

// -----------------------------------------------------------------------------
extern "C" void kernel_launch(void* const* d_in, const int* in_sizes, int n_in,
                              void* d_out, int out_size, void* d_ws, size_t ws_size,
                              hipStream_t stream) {
  const float* x    = (const float*)d_in[0];
  const int*   ei   = (const int*)d_in[1];
  const float* Wz   = (const float*)d_in[2];
  const float* bz   = (const float*)d_in[3];
  // d_in[4], d_in[5] (Wr, br) are dead: H0==0 -> H0*R==0
  const float* Wh   = (const float*)d_in[6];
  const float* bh   = (const float*)d_in[7];
  const float* Lz   = (const float*)d_in[8];   // [256,128]; only rows 0..127 live
  const float* lbz  = (const float*)d_in[9];
  // d_in[10], d_in[11] (Lr, lbr) dead
  const float* Lh   = (const float*)d_in[12];
  const float* lbh  = (const float*)d_in[13];
  const float* Wout = (const float*)d_in[14];
  const float* bout = (const float*)d_in[15];
  float* out = (float*)d_out;

  char* ws = (char*)d_ws;
  const size_t buf_elems = (size_t)N_NODES * HID;
  const size_t dinv_off  = 0;
  const size_t buf0_off  = ((size_t)N_NODES * sizeof(float) + 255) & ~(size_t)255;
  const size_t need = buf0_off + 3 * buf_elems * sizeof(float);
  if (ws_size < need) return;                  // deterministic no-op guard

  float* dinv = (float*)(ws + dinv_off);
  float* buf0 = (float*)(ws + buf0_off);       // h (per-gate, reused)
  float* buf1 = buf0 + buf_elems;              // aggz
  float* buf2 = buf1 + buf_elems;              // aggh

  const int* src = ei;
  const int* dst = ei + N_EDGES;

  // degree -> dinv
  k_init_deg<<<(N_NODES + 255) / 256, 256, 0, stream>>>(dinv);
  k_edge_deg<<<(N_EDGES + 255) / 256, 256, 0, stream>>>(dst, dinv);
  k_rsqrt  <<<(N_NODES + 255) / 256, 256, 0, stream>>>(dinv);

  // gate z: feature GEMM -> self-loop+bias init -> edge scatter
  k_feat_gemm<<<N_NODES / 16, 256, 0, stream>>>(x, Wz, buf0);
  k_agg_init <<<(N_NODES * HID) / 256, 256, 0, stream>>>(buf0, dinv, bz, buf1);
  k_scatter  <<<N_EDGES / 8, 256, 0, stream>>>(src, dst, dinv, buf0, buf1);

  // gate h
  k_feat_gemm<<<N_NODES / 16, 256, 0, stream>>>(x, Wh, buf0);
  k_agg_init <<<(N_NODES * HID) / 256, 256, 0, stream>>>(buf0, dinv, bh, buf2);
  k_scatter  <<<N_EDGES / 8, 256, 0, stream>>>(src, dst, dinv, buf0, buf2);

  // fused gates + GRU combine + output projection
  k_gate_out<<<N_NODES / 16, 256, 0, stream>>>(buf1, buf2, Lz, lbz, Lh, lbh,
                                               Wout, bout, out);
}